// MetaAttentionModule_86955907875077
// MI455X (gfx1250) — compile-verified
//
#include <hip/hip_runtime.h>

// MetaAttention fused kernel for MI455X (gfx1250, wave32).
// B=4096, N=16, E=512, T=128, S=256.  One wave per batch row.
// Phase 1: scores + softmax with VALU + wave shuffles (f32).
// Phase 2: aggregated[b,:] = w(1x16) @ X(16x512) via V_WMMA_F32_16X16X4_F32,
//          two 16-wide e-tiles per iteration so stores are branchless/all-lane.

#define HAS_WMMA_F32X4 __has_builtin(__builtin_amdgcn_wmma_f32_16x16x4_f32)

typedef __attribute__((ext_vector_type(2))) float v2f;
typedef __attribute__((ext_vector_type(8))) float v8f;

__global__ __launch_bounds__(256) void meta_attn_kernel(
    const float* __restrict__ task_anchor,   // (B, 128)
    const float* __restrict__ expert,        // (B, 16, 512)
    const float* __restrict__ v,             // (640, 1): v_e = v[0:512], v_t = v[512:640]
    float* __restrict__ agg,                 // (B, 512)
    float* __restrict__ attw)                // (B, 16)
{
    const int lane = threadIdx.x & 31;
    const int wv   = threadIdx.x >> 5;
    const int b    = blockIdx.x * 8 + wv;
    const int n    = lane & 15;      // expert index this lane owns
    const int half = lane >> 4;      // 0: lanes 0-15, 1: lanes 16-31

    const float* X = expert + (size_t)b * (16 * 512);

    // ---------------- Phase 1: scores[b, n] -----------------
    // Each lane streams a contiguous 256-float chunk of row n (float4 loads).
    float partial = 0.0f;
    {
        const float4* xr = (const float4*)(X + n * 512 + half * 256);
        const float4* vr = (const float4*)(v + half * 256);
#pragma unroll 8
        for (int i = 0; i < 64; ++i) {
            float4 x4 = xr[i];
            float4 v4 = vr[i];
            partial += x4.x * v4.x + x4.y * v4.y + x4.z * v4.z + x4.w * v4.w;
        }
    }
    float score = partial + __shfl_xor(partial, 16, 32);   // full dot over 512

    // Task-anchor dot: 128 elements over 32 lanes (4 each), butterfly-reduce.
    {
        float4 ta = ((const float4*)(task_anchor + (size_t)b * 128))[lane];
        float4 vt = ((const float4*)(v + 512))[lane];
        float t = ta.x * vt.x + ta.y * vt.y + ta.z * vt.z + ta.w * vt.w;
#pragma unroll
        for (int m = 16; m >= 1; m >>= 1) t += __shfl_xor(t, m, 32);
        score += t;
    }

    // ---------------- Softmax over 16 experts ----------------
    // score for expert n lives (duplicated) in lanes n and n+16; butterfly over
    // the low 4 lane bits covers all 16 experts within each half.
    float mx = score;
#pragma unroll
    for (int m = 8; m >= 1; m >>= 1) mx = fmaxf(mx, __shfl_xor(mx, m, 32));
    float ex = __expf(score - mx);
    float s = ex;
#pragma unroll
    for (int m = 8; m >= 1; m >>= 1) s += __shfl_xor(s, m, 32);
    const float w = ex / s;          // attention weight for expert n (all lanes valid)

    if (lane < 16) attw[(size_t)b * 16 + n] = w;

    // ---------------- Phase 2: aggregated = w @ X ----------------
#if HAS_WMMA_F32X4
    // A-matrix (16x4 f32) layout: lane L holds M=L&15; K = vgprIdx + 2*laneGroup.
    // We broadcast w over M: a[kk] covers K = {k0+2h, k0+2h+1}, k0 = 4*kk.
    v2f a[4];
#pragma unroll
    for (int kk = 0; kk < 4; ++kk) {
        const int base = kk * 4 + 2 * half;
        a[kk].x = __shfl(w, base, 32);
        a[kk].y = __shfl(w, base + 1, 32);
    }

    // B-matrix (4x16 f32) layout: row K = vgprIdx + 2*laneGroup, col N = lane&15.
    const float* Bbase = X + (2 * half) * 512 + n;
    float* aggr = agg + (size_t)b * 512;

    // Two 16-wide e-tiles per iteration: lanes 0-15 emit tile0 (cols e0+0..15),
    // lanes 16-31 emit tile1 (cols e0+16..31) -> branchless all-lane store.
    for (int e0 = 0; e0 < 512; e0 += 32) {
        v8f c0 = {};
        v8f c1 = {};
#pragma unroll
        for (int kk = 0; kk < 4; ++kk) {
            v2f bm0, bm1;
            bm0.x = Bbase[(kk * 4 + 0) * 512 + e0];        // row k0 + 2h + 0
            bm0.y = Bbase[(kk * 4 + 1) * 512 + e0];        // row k0 + 2h + 1
            bm1.x = Bbase[(kk * 4 + 0) * 512 + e0 + 16];
            bm1.y = Bbase[(kk * 4 + 1) * 512 + e0 + 16];
            // (neg_a, A, neg_b, B, c_mod, C, reuse_a, reuse_b)
            c0 = __builtin_amdgcn_wmma_f32_16x16x4_f32(
                     false, a[kk], false, bm0, (short)0, c0, false, false);
            c1 = __builtin_amdgcn_wmma_f32_16x16x4_f32(
                     false, a[kk], false, bm1, (short)0, c1, false, false);
        }
        // D rows are identical (A broadcast over M). VGPR0 at lane l holds
        // D[., N=l&15]; select tile by lane half and store column e0+lane.
        const float val = half ? c1[0] : c0[0];
        aggr[e0 + lane] = val;
    }
#else
    // VALU fallback (keeps the build green if the f32 WMMA builtin is absent).
    float w16[16];
#pragma unroll
    for (int k = 0; k < 16; ++k) w16[k] = __shfl(w, k, 32);
    float* aggr = agg + (size_t)b * 512;
    for (int e0 = 0; e0 < 512; e0 += 32) {
        float acc = 0.0f;
#pragma unroll
        for (int k = 0; k < 16; ++k) acc += w16[k] * X[k * 512 + e0 + lane];
        aggr[e0 + lane] = acc;
    }
#endif
}

extern "C" void kernel_launch(void* const* d_in, const int* in_sizes, int n_in,
                              void* d_out, int out_size, void* d_ws, size_t ws_size,
                              hipStream_t stream) {
    (void)in_sizes; (void)n_in; (void)d_ws; (void)ws_size; (void)out_size;
    // setup_inputs order: scene_repr (unused), task_anchor, expert_reprs, v
    const float* task_anchor = (const float*)d_in[1];
    const float* expert      = (const float*)d_in[2];
    const float* v           = (const float*)d_in[3];

    float* agg  = (float*)d_out;                       // (4096, 512)
    float* attw = (float*)d_out + (size_t)4096 * 512;  // (4096, 16)

    // 1 wave per batch row, 8 waves (256 threads) per block -> 512 blocks.
    meta_attn_kernel<<<dim3(4096 / 8), dim3(256), 0, stream>>>(
        task_anchor, expert, v, agg, attw);
}